// CategoryAwareDAHEAD_23493471109707
// MI455X (gfx1250) — compile-verified
//
#include <hip/hip_runtime.h>
#include <hip/hip_bf16.h>
#include <math.h>

// ---------------- problem constants (from reference) ----------------
#define NTOT 512
#define NSRC 256
#define NC   21
#define DF   2048
#define H1D  1024
#define DIN  2069           // DF + NC
#define DPAD 2080           // DIN padded to multiple of 4 (128B-aligned rows)
#define MOMc 0.5f
#define THR0c 0.1f
#define EPSc 1e-8f
#define LNEPS 1e-5f

typedef float v2f __attribute__((ext_vector_type(2)));
typedef float v8f __attribute__((ext_vector_type(8)));

// ---------------- workspace layout (float offsets) ----------------
#define WS_OP    0                                  // 512*2080
#define WS_HA    (WS_OP  + NTOT*DPAD)               // 512*1024
#define WS_HB    (WS_HA  + NTOT*H1D)                // 512*1024
#define WS_PROT  (WS_HB  + NTOT*H1D)                // 2*21*2048
#define WS_NRM   (WS_PROT + 2*NC*DF)                // 64 (2*21 norms, clamped)
#define WS_BWL   (WS_NRM + 64)                      // 21*2048
#define WS_BN    (WS_BWL + NC*DF)                   // 32 (bwl norms, clamped)
#define WS_CNT   (WS_BN  + 32)                      // 32 (per-class src counts)
#define WS_MAXP  (WS_CNT + 32)                      // 512
#define WS_MAXC  (WS_MAXP + NTOT)                   // 512 ints
#define WS_ACC   (WS_MAXC + NTOT)                   // [0]=rm_sum, [1]=ea_sum

__device__ __forceinline__ float wave_sum(float v) {
#pragma unroll
  for (int off = 16; off > 0; off >>= 1) v += __shfl_down(v, off, 32);
  return v;
}

// ---------------- k0: zero the loss accumulators ----------------
__global__ void k_init(float* ws) {
  ws[WS_ACC + 0] = 0.f;
  ws[WS_ACC + 1] = 0.f;
}

// ---------------- k1: 7x7 mean pool + concat logits + zero pad -> op ----------------
__global__ void k_pool_concat(const float* __restrict__ feats,
                              const float* __restrict__ logits,
                              float* __restrict__ op) {
  long tid = (long)blockIdx.x * blockDim.x + threadIdx.x;
  if (tid >= (long)NTOT * DPAD) return;
  int n = (int)(tid / DPAD);
  int c = (int)(tid % DPAD);
  float v;
  if (c < DF) {
    const float* p = feats + ((long)n * DF + c) * 49;
    __builtin_prefetch(p, 0, 0);               // global_prefetch_b8
    float s = 0.f;
#pragma unroll
    for (int i = 0; i < 49; ++i) s += p[i];
    v = s / 49.f;
  } else if (c < DIN) {
    v = logits[n * NC + (c - DF)];
  } else {
    v = 0.f;
  }
  op[(long)n * DPAD + c] = v;
}

// ---------------- k2: softmax max prob + argmax per sample ----------------
__global__ void k_softmax_max(const float* __restrict__ logits,
                              float* __restrict__ maxp, int* __restrict__ maxc) {
  int n = blockIdx.x * blockDim.x + threadIdx.x;
  if (n >= NTOT) return;
  const float* l = logits + n * NC;
  float m = l[0]; int am = 0;
#pragma unroll
  for (int i = 1; i < NC; ++i) { float v = l[i]; if (v > m) { m = v; am = i; } }
  float s = 0.f;
#pragma unroll
  for (int i = 0; i < NC; ++i) s += expf(l[i] - m);
  maxp[n] = 1.f / s;          // max of softmax == exp(0)/sum
  maxc[n] = am;
}

// ---------------- k3: sequential threshold-gated prototype scan ----------------
// grid = 42 blocks: blockIdx = half*21 + cls ; 256 threads x 8 dims
__global__ void k_proto_scan(const float* __restrict__ op,
                             const float* __restrict__ maxp,
                             const int* __restrict__ maxc,
                             float* __restrict__ protos, float* __restrict__ norms) {
  int half = blockIdx.x / NC;
  int cls  = blockIdx.x % NC;
  int t = threadIdx.x;
  float p[8];
#pragma unroll
  for (int j = 0; j < 8; ++j) p[j] = 0.f;
  float thresh = THR0c;
  int step = 0;
  for (int i = 0; i < NSRC; ++i) {
    int idx = half * NSRC + i;
    if (maxc[idx] == cls) {
      float ml = maxp[idx];
      if (ml >= thresh) {
        step += 1;
        float fs = (float)step;
        const float* x = op + (long)idx * DPAD;
#pragma unroll
        for (int j = 0; j < 8; ++j) p[j] = (p[j] + x[t + j * 256]) / fs;
        thresh = thresh * MOMc + ml * (1.f - MOMc);
      }
    }
  }
  float* out = protos + ((long)(half * NC + cls)) * DF;
  float q = 0.f;
#pragma unroll
  for (int j = 0; j < 8; ++j) { out[t + j * 256] = p[j]; q += p[j] * p[j]; }
  __shared__ float red[256];
  red[t] = q; __syncthreads();
  for (int off = 128; off > 0; off >>= 1) { if (t < off) red[t] += red[t + off]; __syncthreads(); }
  if (t == 0) norms[half * NC + cls] = fmaxf(sqrtf(red[0]), EPSc);
}

// ---------------- k4: per-class mean of src_mapped (bwl) ----------------
__global__ void k_bwl(const float* __restrict__ op, const int* __restrict__ labels,
                      float* __restrict__ bwl, float* __restrict__ bnorm,
                      float* __restrict__ counts) {
  int cls = blockIdx.x;
  int t = threadIdx.x;
  float s[8];
#pragma unroll
  for (int j = 0; j < 8; ++j) s[j] = 0.f;
  int cnt = 0;
  for (int i = 0; i < NSRC; ++i) {
    if (labels[i] == cls) {
      ++cnt;
      const float* x = op + (long)i * DPAD;
#pragma unroll
      for (int j = 0; j < 8; ++j) s[j] += x[t + j * 256];
    }
  }
  float q = 0.f;
  float* out = bwl + (long)cls * DF;
#pragma unroll
  for (int j = 0; j < 8; ++j) {
    float v = (cnt > 0) ? (s[j] / (float)cnt) : 0.f;
    out[t + j * 256] = v; q += v * v;
  }
  __shared__ float red[256];
  red[t] = q; __syncthreads();
  for (int off = 128; off > 0; off >>= 1) { if (t < off) red[t] += red[t + off]; __syncthreads(); }
  if (t == 0) { bnorm[cls] = fmaxf(sqrtf(red[0]), EPSc); counts[cls] = (float)cnt; }
}

// ---------------- k5: 20x20 masked |cos - cos| sum -> rm accumulator ----------------
__global__ void k_cos_rm(const float* __restrict__ tpro, const float* __restrict__ spro,
                         const float* __restrict__ bwl,
                         const float* __restrict__ tn, const float* __restrict__ sn,
                         const float* __restrict__ bn, const float* __restrict__ counts,
                         float* __restrict__ acc) {
  int wave = (blockIdx.x * blockDim.x + threadIdx.x) >> 5;
  int lane = threadIdx.x & 31;
  if (wave >= 400) return;
  int i = wave / 20, j = wave % 20;            // t class i+1, s class j+1
  const float* T = tpro + (long)(i + 1) * DF;
  const float* S = spro + (long)(j + 1) * DF;
  const float* B = bwl  + (long)(j + 1) * DF;
  float d1 = 0.f, d2 = 0.f;
  for (int k = lane; k < DF; k += 32) { float tv = T[k]; d1 += tv * S[k]; d2 += tv * B[k]; }
  d1 = wave_sum(d1); d2 = wave_sum(d2);
  if (lane == 0) {
    float ptm    = d1 / (tn[i + 1] * sn[j + 1]);
    float bwltpm = d2 / (tn[i + 1] * bn[j + 1]);
    float mask = (counts[i + 1] > 0.f) ? 1.f : 0.f;
    atomicAdd(acc, fabsf(bwltpm - ptm) * mask);
  }
}

// ---------------- k6: WMMA f32 GEMM: C[M,N] = A[M,K] * B[K,N] + bias ----------------
// B/C have compile-time leading dimension H1D. Each wave computes a 16(M) x
// 64(N) tile: 4 accumulators share one A fragment per K-step of 4
// (V_WMMA_F32_16X16X4_F32). Running pointers -> B loads are immediate-offset
// from one base; branch-free main loop, single guarded tail step (A rows are
// zero-padded past kvalid so tail products stay exact, B never read OOB).
__global__ void k_gemm_wmma(const float* __restrict__ A, int lda,
                            const float* __restrict__ B, int kvalid,
                            const float* __restrict__ bias,
                            float* __restrict__ C, int tilesN4) {
  int tile = blockIdx.x * (blockDim.x >> 5) + (threadIdx.x >> 5);
  int tm = tile / tilesN4, tn4 = tile % tilesN4;
  int lane = threadIdx.x & 31;
  int hi   = lane >> 4;            // 0: K sub 0/1, 1: K sub 2/3
  int l15  = lane & 15;
  int arow  = tm * 16 + l15;
  int bcol  = tn4 * 64 + l15;
  int ko    = hi * 2;
  const float* Ap = A + (long)arow * lda + ko;        // += 4 per step
  const float* Bp = B + (long)ko * H1D + bcol;        // += 4*H1D per step
  v8f a0 = {}, a1 = {}, a2 = {}, a3 = {};
  int kfull = kvalid & ~3;
  for (int k0 = 0; k0 < kfull; k0 += 4) {
    v2f a; a.x = Ap[0]; a.y = Ap[1];
    v2f b0, b1, b2, b3;
    b0.x = Bp[0];  b0.y = Bp[H1D];
    b1.x = Bp[16]; b1.y = Bp[H1D + 16];
    b2.x = Bp[32]; b2.y = Bp[H1D + 32];
    b3.x = Bp[48]; b3.y = Bp[H1D + 48];
    a0 = __builtin_amdgcn_wmma_f32_16x16x4_f32(false, a, false, b0, (short)0, a0, false, false);
    a1 = __builtin_amdgcn_wmma_f32_16x16x4_f32(false, a, false, b1, (short)0, a1, false, false);
    a2 = __builtin_amdgcn_wmma_f32_16x16x4_f32(false, a, false, b2, (short)0, a2, false, false);
    a3 = __builtin_amdgcn_wmma_f32_16x16x4_f32(false, a, false, b3, (short)0, a3, false, false);
    Ap += 4;
    Bp += 4 * H1D;
  }
  if (kfull < kvalid) {            // tail (GEMM1 only: K=2069 -> 1 live column)
    int ka = kfull + ko;
    v2f a; a.x = Ap[0]; a.y = Ap[1];              // zero-padded, always safe
    bool v0 = (ka < kvalid), v1 = (ka + 1 < kvalid);
    v2f b0, b1, b2, b3;
    b0.x = v0 ? Bp[0]  : 0.f;  b0.y = v1 ? Bp[H1D]      : 0.f;
    b1.x = v0 ? Bp[16] : 0.f;  b1.y = v1 ? Bp[H1D + 16] : 0.f;
    b2.x = v0 ? Bp[32] : 0.f;  b2.y = v1 ? Bp[H1D + 32] : 0.f;
    b3.x = v0 ? Bp[48] : 0.f;  b3.y = v1 ? Bp[H1D + 48] : 0.f;
    a0 = __builtin_amdgcn_wmma_f32_16x16x4_f32(false, a, false, b0, (short)0, a0, false, false);
    a1 = __builtin_amdgcn_wmma_f32_16x16x4_f32(false, a, false, b1, (short)0, a1, false, false);
    a2 = __builtin_amdgcn_wmma_f32_16x16x4_f32(false, a, false, b2, (short)0, a2, false, false);
    a3 = __builtin_amdgcn_wmma_f32_16x16x4_f32(false, a, false, b3, (short)0, a3, false, false);
  }
  // D layout: VGPR r <-> row = r + 8*hi ; N = l15 within each 16-wide subtile
  float* Cp = C + (long)(tm * 16 + hi * 8) * H1D + bcol;
#pragma unroll
  for (int s = 0; s < 4; ++s) {
    int col = 16 * s;
    float bb = bias[bcol + col];
    v8f* accp = (s == 0) ? &a0 : (s == 1) ? &a1 : (s == 2) ? &a2 : &a3;
#pragma unroll
    for (int r = 0; r < 8; ++r) {
      Cp[(long)r * H1D + col] = (*accp)[r] + bb;
    }
  }
}

// ---------------- k7: in-place LayerNorm + ReLU over rows of 1024 ----------------
__global__ void k_ln_relu(float* __restrict__ H) {
  int row = blockIdx.x;
  int t = threadIdx.x;                         // 256 threads x 4 elems
  float* h = H + (long)row * H1D;
  float v[4]; float s = 0.f;
#pragma unroll
  for (int j = 0; j < 4; ++j) { v[j] = h[t + j * 256]; s += v[j]; }
  __shared__ float red[256];
  red[t] = s; __syncthreads();
  for (int off = 128; off > 0; off >>= 1) { if (t < off) red[t] += red[t + off]; __syncthreads(); }
  float mu = red[0] / (float)H1D;
  __syncthreads();
  float q = 0.f;
#pragma unroll
  for (int j = 0; j < 4; ++j) { float d = v[j] - mu; q += d * d; }
  red[t] = q; __syncthreads();
  for (int off = 128; off > 0; off >>= 1) { if (t < off) red[t] += red[t + off]; __syncthreads(); }
  float inv = rsqrtf(red[0] / (float)H1D + LNEPS);
#pragma unroll
  for (int j = 0; j < 4; ++j) h[t + j * 256] = fmaxf((v[j] - mu) * inv, 0.f);
}

// ---------------- k8: final dot + sigmoid + BCE-style term -> ea accumulator ----------------
__global__ void k_head_ea(const float* __restrict__ H, const float* __restrict__ Wd,
                          const float* __restrict__ bd, const int* __restrict__ dl,
                          float* __restrict__ acc) {
  int wave = (blockIdx.x * blockDim.x + threadIdx.x) >> 5;
  int lane = threadIdx.x & 31;
  if (wave >= NTOT) return;
  const float* h = H + (long)wave * H1D;
  float d = 0.f;
  for (int k = lane; k < H1D; k += 32) d += h[k] * Wd[k];
  d = wave_sum(d);
  if (lane == 0) {
    float z = 1.f / (1.f + expf(-(d + bd[0])));
    float y = (float)dl[wave];
    float v = fmaxf(z, 0.f) - z * y + log1pf(expf(-fabsf(z)));
    atomicAdd(acc, v);
  }
}

// ---------------- k9: finalize both losses ----------------
__global__ void k_finalize(const float* __restrict__ counts,
                           const float* __restrict__ acc, float* __restrict__ out) {
  float msum = 0.f;
  for (int i = 1; i < NC; ++i) msum += (counts[i] > 0.f) ? 1.f : 0.f;
  out[0] = 1.0f * acc[0] / (msum * (float)(NC - 1));   // MGRM_W = 1.0
  out[1] = acc[1] / (float)NTOT;
}

extern "C" void kernel_launch(void* const* d_in, const int* in_sizes, int n_in,
                              void* d_out, int out_size, void* d_ws, size_t ws_size,
                              hipStream_t stream) {
  const float* feats  = (const float*)d_in[0];   // (512,2048,7,7)
  const float* logits = (const float*)d_in[1];   // (512,21)
  const float* W1 = (const float*)d_in[2];  const float* b1 = (const float*)d_in[3];
  const float* W2 = (const float*)d_in[4];  const float* b2 = (const float*)d_in[5];
  const float* W3 = (const float*)d_in[6];  const float* b3 = (const float*)d_in[7];
  const float* Wd = (const float*)d_in[8];  const float* bd = (const float*)d_in[9];
  const int* labels = (const int*)d_in[10];
  const int* dl     = (const int*)d_in[11];
  float* out = (float*)d_out;

  float* ws = (float*)d_ws;
  float* op   = ws + WS_OP;
  float* hA   = ws + WS_HA;
  float* hB   = ws + WS_HB;
  float* prot = ws + WS_PROT;      // half 0 = src protos, half 1 = tgt protos
  float* nrm  = ws + WS_NRM;
  float* bwl  = ws + WS_BWL;
  float* bn   = ws + WS_BN;
  float* cnt  = ws + WS_CNT;
  float* maxp = ws + WS_MAXP;
  int*   maxc = (int*)(ws + WS_MAXC);
  float* acc  = ws + WS_ACC;

  k_init<<<1, 1, 0, stream>>>(ws);

  { long tot = (long)NTOT * DPAD;
    k_pool_concat<<<(int)((tot + 255) / 256), 256, 0, stream>>>(feats, logits, op); }

  k_softmax_max<<<2, 256, 0, stream>>>(logits, maxp, maxc);

  k_proto_scan<<<2 * NC, 256, 0, stream>>>(op, maxp, maxc, prot, nrm);

  k_bwl<<<NC, 256, 0, stream>>>(op, labels, bwl, bn, cnt);

  // loss_rm: t = half 1, s = half 0
  k_cos_rm<<<50, 256, 0, stream>>>(prot + (long)NC * DF, prot, bwl,
                                   nrm + NC, nrm, bn, cnt, acc);

  // MLP GEMMs: waves of 16x64 tiles => (512/16)*(1024/64) = 512 waves, 4/block
  const int gemmBlocks = (NTOT / 16) * (H1D / 64) / 4;
  k_gemm_wmma<<<gemmBlocks, 128, 0, stream>>>(op, DPAD, W1, DIN, b1,
                                              hA, H1D / 64);
  k_ln_relu<<<NTOT, 256, 0, stream>>>(hA);

  k_gemm_wmma<<<gemmBlocks, 128, 0, stream>>>(hA, H1D, W2, H1D, b2,
                                              hB, H1D / 64);
  k_ln_relu<<<NTOT, 256, 0, stream>>>(hB);

  k_gemm_wmma<<<gemmBlocks, 128, 0, stream>>>(hB, H1D, W3, H1D, b3,
                                              hA, H1D / 64);
  k_ln_relu<<<NTOT, 256, 0, stream>>>(hA);

  k_head_ea<<<(NTOT * 32) / 256, 256, 0, stream>>>(hA, Wd, bd, dl, acc);

  k_finalize<<<1, 1, 0, stream>>>(cnt, acc, out);
}